// PatchedEnergyScore_74938589381140
// MI455X (gfx1250) — compile-verified
//
#include <hip/hip_runtime.h>
#include <hip/hip_bf16.h>
#include <math.h>

// PatchedEnergyScore on MI455X (gfx1250), wave32 + WMMA f32 16x16x4.
// Pipeline: stage1 per-column 16x16 Grams (WMMA, two columns per iteration via
// b64 loads) -> stage2 mask GEMM (WMMA, both p-tiles fused per wave, K-slab
// partials) -> deterministic slab reduce -> tiny finalize (sqrt/score).
// Assumes m=16, v=16, P=32, L even (holds for the reference setup).

typedef __attribute__((ext_vector_type(2))) float v2f;
typedef __attribute__((ext_vector_type(8))) float v8f;

#define NB_PER_B 296   // per-batch column block: 256 PP + 32 PT-halves + 2 TT-halves + 6 pad

// ---------------------------------------------------------------------------
// Stage 1: for each (b, l) compute the 16x16 Gram over v of p_hat = preds*s_v*w_l,
// plus PT halves and TT halves. One wave streams a strip of l values, two
// columns at a time (adjacent l are adjacent in memory -> 64-bit loads).
// A and B register contents are identical for a symmetric Gram.
// ---------------------------------------------------------------------------
__global__ __launch_bounds__(32)
void stage1_colgram(const float* __restrict__ preds,
                    const float* __restrict__ target,
                    const float* __restrict__ weights,
                    const float* __restrict__ scale,
                    float* __restrict__ C,
                    int L, int stripLen, int nstride)
{
    const int b  = blockIdx.y;
    const int ln = threadIdx.x;      // 0..31
    const int r  = ln & 15;          // row (A) / col (B) index
    const int hi = ln >> 4;          // K-half select

    // Per-lane v indices: chunk vb covers v = 4*vb .. 4*vb+3; this lane owns
    // v = 4*vb + 2*hi + {0,1}. Preload scale factors and flat offsets.
    float  sv[8];
    size_t offP[8], offT[8];
#pragma unroll
    for (int vb = 0; vb < 4; ++vb) {
#pragma unroll
        for (int k = 0; k < 2; ++k) {
            const int vv = vb * 4 + 2 * hi + k;
            sv[vb * 2 + k]   = scale[vv];
            offP[vb * 2 + k] = (size_t)((b * 16 + r) * 16 + vv) * (size_t)L;
            offT[vb * 2 + k] = (size_t)(b * 16 + vv) * (size_t)L;
        }
    }

    const int l0   = blockIdx.x * stripLen;
    const int lend = min(l0 + stripLen, L);

    for (int l = l0; l < lend; l += 2) {
        const float w0 = weights[l];
        const float w1 = weights[l + 1];
        v8f acc0 = {0.f, 0.f, 0.f, 0.f, 0.f, 0.f, 0.f, 0.f};
        v8f acc1 = {0.f, 0.f, 0.f, 0.f, 0.f, 0.f, 0.f, 0.f};
        float pt0 = 0.f, tt0 = 0.f, pt1 = 0.f, tt1 = 0.f;
#pragma unroll
        for (int vb = 0; vb < 4; ++vb) {
            // 64-bit loads: .x belongs to column l, .y to column l+1
            const v2f p0 = *(const v2f*)(preds  + offP[vb * 2 + 0] + l);
            const v2f p1 = *(const v2f*)(preds  + offP[vb * 2 + 1] + l);
            const v2f q0 = *(const v2f*)(target + offT[vb * 2 + 0] + l);
            const v2f q1 = *(const v2f*)(target + offT[vb * 2 + 1] + l);
            const float s0 = sv[vb * 2 + 0], s1 = sv[vb * 2 + 1];

            v2f A0; A0.x = p0.x * (s0 * w0); A0.y = p1.x * (s1 * w0);  // column l
            v2f A1; A1.x = p0.y * (s0 * w1); A1.y = p1.y * (s1 * w1);  // column l+1
            acc0 = __builtin_amdgcn_wmma_f32_16x16x4_f32(
                false, A0, false, A0, (short)0, acc0, false, false);
            acc1 = __builtin_amdgcn_wmma_f32_16x16x4_f32(
                false, A1, false, A1, (short)0, acc1, false, false);

            const float u0 = q0.x * (s0 * w0), u1 = q1.x * (s1 * w0);  // t_hat, col l
            const float x0 = q0.y * (s0 * w1), x1 = q1.y * (s1 * w1);  // t_hat, col l+1
            pt0 += A0.x * u0 + A0.y * u1;
            tt0 += u0 * u0 + u1 * u1;
            pt1 += A1.x * x0 + A1.y * x1;
            tt1 += x0 * x0 + x1 * x1;
        }
        float* Crow0 = C + (size_t)l * nstride + b * NB_PER_B;
        float* Crow1 = Crow0 + nstride;
#pragma unroll
        for (int e = 0; e < 8; ++e) {            // D element e -> (M=e+8*hi, N=r)
            Crow0[(e + 8 * hi) * 16 + r] = acc0[e];
            Crow1[(e + 8 * hi) * 16 + r] = acc1[e];
        }
        Crow0[256 + ln] = pt0;                   // PT halves: PT[i] = [256+i] + [272+i]
        Crow1[256 + ln] = pt1;
        if (r == 0) {                            // TT halves: TT = [288] + [289]
            Crow0[288 + hi] = tt0;
            Crow1[288 + hi] = tt1;
        }
    }
}

// ---------------------------------------------------------------------------
// Stage 2: G[p,n] = sum_l masks[p,l] * C[l,n].  M=32 (both 16-row tiles fused
// per wave so C is read once), N=592 (37 tiles), K=L in slabs of Klen. Each
// wave writes its two 16x16 slab-partial tiles (no atomics, deterministic).
// ---------------------------------------------------------------------------
__global__ __launch_bounds__(32)
void stage2_maskgemm(const float* __restrict__ masks,
                     const float* __restrict__ C,
                     float* __restrict__ Gpart,
                     int L, int Klen, int nstride, int Prows)
{
    const int ln = threadIdx.x;
    const int r  = ln & 15;
    const int hi = ln >> 4;
    const int n  = blockIdx.y * 16 + r;                  // B column / D column
    const float* Arow0 = masks + (size_t)r * L;          // mask row r       (tile 0)
    const float* Arow1 = masks + (size_t)(16 + r) * L;   // mask row 16 + r  (tile 1)

    const int l0   = blockIdx.x * Klen;
    const int lend = min(l0 + Klen, L);

    v8f acc0 = {0.f, 0.f, 0.f, 0.f, 0.f, 0.f, 0.f, 0.f};
    v8f acc1 = {0.f, 0.f, 0.f, 0.f, 0.f, 0.f, 0.f, 0.f};
    for (int l = l0; l < lend; l += 4) {
        // A pairs are adjacent in l -> 8-byte-aligned 64-bit loads
        const v2f a0 = *(const v2f*)(Arow0 + l + 2 * hi);
        const v2f a1 = *(const v2f*)(Arow1 + l + 2 * hi);
        v2f bb;
        bb.x = C[(size_t)(l + 2 * hi)     * nstride + n]; // B: col n, K = 2*hi+{0,1}
        bb.y = C[(size_t)(l + 2 * hi + 1) * nstride + n];
        acc0 = __builtin_amdgcn_wmma_f32_16x16x4_f32(
            false, a0, false, bb, (short)0, acc0, false, false);
        acc1 = __builtin_amdgcn_wmma_f32_16x16x4_f32(
            false, a1, false, bb, (short)0, acc1, false, false);
    }

    float* g = Gpart + (size_t)blockIdx.x * Prows * (size_t)nstride;
#pragma unroll
    for (int e = 0; e < 8; ++e) {
        g[(size_t)(e + 8 * hi)      * nstride + blockIdx.y * 16 + r] = acc0[e];
        g[(size_t)(16 + e + 8 * hi) * nstride + blockIdx.y * 16 + r] = acc1[e];
    }
}

// Deterministic reduction of K-slab partials: G[i] = sum_s Gpart[s][i].
__global__ void reduce_g(const float* __restrict__ Gpart, float* __restrict__ G,
                         int S2, int totalPN)
{
    const int i = blockIdx.x * blockDim.x + threadIdx.x;
    if (i >= totalPN) return;
    float s = 0.f;
    for (int k = 0; k < S2; ++k) s += Gpart[(size_t)k * totalPN + i];
    G[i] = s;
}

// ---------------------------------------------------------------------------
// Finalize: per (p,b) read the 17x17 Gram pieces and compute the energy score
// terms; single-block reduction (deterministic).
// ---------------------------------------------------------------------------
__global__ void finalize_k(const float* __restrict__ G, const int* __restrict__ beta_p,
                           float* __restrict__ out, int m, int bs, int P, int nstride)
{
    __shared__ float red[256];
    const int t = threadIdx.x;
    float val = 0.f;
    if (t < P * bs) {
        const int p = t / bs;
        const int b = t % bs;
        const float* g   = G + (size_t)p * nstride + b * NB_PER_B;
        const float beta = (float)(*beta_p);
        const float TT   = g[288] + g[289];
        float sa = 0.f, sb = 0.f;
        for (int i = 0; i < m; ++i) {
            const float PTi = g[256 + i] + g[272 + i];
            const float d2  = g[i * 16 + i] - 2.f * PTi + TT;   // |x_i - y|^2
            const float a   = sqrtf(fmaxf(d2, 0.f));
            sa += (beta == 1.f) ? a : powf(a, beta);
        }
        for (int i = 0; i < m; ++i) {
            const float Pii = g[i * 16 + i];
            for (int j = 0; j < m; ++j) {
                const float d2 = Pii + g[j * 16 + j] - 2.f * g[i * 16 + j];
                const float d  = (d2 > 0.f) ? sqrtf(d2) : 0.f;
                sb += (beta == 1.f) ? d : powf(d, beta);
            }
        }
        val = sa / ((float)bs * (float)m)
            - sb / ((float)bs * 2.f * (float)m * (float)(m - 1));
    }
    red[t] = val;
    __syncthreads();
    if (t == 0) {
        float tot = 0.f;
        for (int i = 0; i < (int)blockDim.x; ++i) tot += red[i];
        out[0] = tot;
    }
}

// ---------------------------------------------------------------------------
extern "C" void kernel_launch(void* const* d_in, const int* in_sizes, int n_in,
                              void* d_out, int out_size, void* d_ws, size_t ws_size,
                              hipStream_t stream)
{
    const float* preds   = (const float*)d_in[0];
    const float* target  = (const float*)d_in[1];
    const float* weights = (const float*)d_in[2];
    const float* scale   = (const float*)d_in[3];
    const float* masks   = (const float*)d_in[4];
    const int*   beta    = (const int*)d_in[5];

    const int L  = in_sizes[2];                 // 40320
    const int v  = in_sizes[3];                 // 16 (layout requires 16)
    const int P  = in_sizes[4] / L;             // 32 (stage2 requires 32)
    const int bs = in_sizes[1] / (v * L);       // 2
    const int m  = in_sizes[0] / (bs * v * L);  // 16 (layout requires 16)

    const int nstride = bs * NB_PER_B;          // 592, multiple of 16 for bs=2
    const int ntiles  = nstride / 16;           // 37

    // Workspace carve-up: C (L x nstride), Gpart (S2 x P x nstride), G (P x nstride)
    const int Klen = 1440;
    const int S2   = (L + Klen - 1) / Klen;     // 28
    float* C = (float*)d_ws;
    size_t cBytes = ((size_t)L * nstride * sizeof(float) + 255) & ~(size_t)255;
    float* Gpart = (float*)((char*)d_ws + cBytes);
    size_t gpBytes = ((size_t)S2 * P * nstride * sizeof(float) + 255) & ~(size_t)255;
    float* G = (float*)((char*)d_ws + cBytes + gpBytes);

    const int stripLen = 32;
    const int strips   = (L + stripLen - 1) / stripLen;

    hipLaunchKernelGGL(stage1_colgram, dim3(strips, bs), dim3(32), 0, stream,
                       preds, target, weights, scale, C, L, stripLen, nstride);
    hipLaunchKernelGGL(stage2_maskgemm, dim3(S2, ntiles), dim3(32), 0, stream,
                       masks, C, Gpart, L, Klen, nstride, P);
    const int totalPN = P * nstride;
    hipLaunchKernelGGL(reduce_g, dim3((totalPN + 255) / 256), dim3(256), 0, stream,
                       Gpart, G, S2, totalPN);
    hipLaunchKernelGGL(finalize_k, dim3(1), dim3(64), 0, stream,
                       G, beta, (float*)d_out, m, bs, P, nstride);
}